// CaptionModel_88003879895249
// MI455X (gfx1250) — compile-verified
//
#include <hip/hip_runtime.h>
#include <hip/hip_bf16.h>
#include <math.h>

#define SEQ_LEN 20
#define BEAM 16
#define VOCAB 50258
#define RNN 1024
#define EMB 1024
#define GATES 4096
#define NEGV (-1e10f)
#define LSTRIDE 50272            // 16 * 3142 (padded vocab)
#define NTILES_OUT 3142
#define NTILES_G 256

typedef float v2f __attribute__((ext_vector_type(2)));
typedef float v8f __attribute__((ext_vector_type(8)));

// ---------------- workspace layout (float units) ----------------
#define OFF_LOGITS   0u                         // 16*LSTRIDE = 804352
#define OFF_A        804352u                    // 16*2048
#define OFF_G        837120u                    // 16*4096
#define OFF_H        902656u                    // 16*1024
#define OFF_C        919040u                    // 16*1024
#define OFF_CG       935424u                    // 16*1024
#define OFF_LPSUM    951808u                    // 16
#define OFF_CVAL     951840u                    // 256
#define OFF_CLPF     952096u                    // 256
#define OFF_CTOK     952352u                    // 256 (int)
#define OFF_SEQ0     952608u                    // 320 (int)
#define OFF_SEQ1     952928u                    // 320 (int)
#define OFF_SLP0     953248u                    // 320
#define OFF_SLP1     953568u                    // 320
#define OFF_SNAPSEQ  953920u                    // 320*20 (int)
#define OFF_SNAPLP   960320u                    // 320*20
#define OFF_SNAPP    966720u                    // 320

__device__ __forceinline__ float sigm(float x) { return 1.0f / (1.0f + expf(-x)); }

// ---------------- init: h,c from inputs; lp_sum = 0 ----------------
__global__ void k_init(const float* __restrict__ ih, const float* __restrict__ ic,
                       float* __restrict__ h, float* __restrict__ c,
                       float* __restrict__ lp_sum) {
  int i = blockIdx.x * blockDim.x + threadIdx.x;   // 16384 threads
  h[i] = ih[i];
  c[i] = ic[i];
  if (i < BEAM) lp_sum[i] = 0.0f;
}

// ---- per-beam: logsumexp (t>0) + log-softmax + constraints + top-16, one block per beam ----
__global__ __launch_bounds__(256) void k_topk(
    const float* __restrict__ src, int stride,
    const int* __restrict__ seq_rd, const float* __restrict__ lp_sum, int t,
    float* __restrict__ cv, float* __restrict__ cl, int* __restrict__ ctk) {
  int b = blockIdx.x, tid = threadIdx.x;
  const float* row = src + (size_t)b * (size_t)stride;

  __shared__ float red[256];
  __shared__ float sls;

  // --- logsumexp over raw logits (only for t>0; init_logprobs already normalized) ---
  if (t > 0) {
    float mx = -3.0e38f;
    for (int v = tid; v < VOCAB; v += 256) mx = fmaxf(mx, row[v]);
    red[tid] = mx; __syncthreads();
    for (int s = 128; s >= 1; s >>= 1) {
      if (tid < s) red[tid] = fmaxf(red[tid], red[tid + s]);
      __syncthreads();
    }
    float m = red[0]; __syncthreads();
    float sum = 0.0f;
    for (int v = tid; v < VOCAB; v += 256) sum += expf(row[v] - m);
    red[tid] = sum; __syncthreads();
    for (int s = 128; s >= 1; s >>= 1) {
      if (tid < s) red[tid] += red[tid + s];
      __syncthreads();
    }
    if (tid == 0) sls = m + logf(red[0]);
    __syncthreads();
  } else if (tid == 0) {
    sls = 0.0f;
  }
  if (t == 0) __syncthreads();

  float base = lp_sum[b];
  float ls = sls;
  int prev = (t > 0) ? seq_rd[(t - 1) * BEAM + b] : -1;
  bool dead0 = (t == 0) && (b > 0);

  float val[16], lf[16]; int tk[16];
#pragma unroll
  for (int j = 0; j < 16; ++j) { val[j] = -3.0e38f; lf[j] = 0.0f; tk[j] = 0; }

  for (int v = tid; v < VOCAB; v += 256) {
    float lp = row[v] - ls;
    if (v == prev) lp = NEGV;
    if (v == VOCAB - 1) lp += -1000.0f;
    float s = dead0 ? NEGV : (base + lp);
    if (s > val[15]) {
      float cs = s, cf = lp; int ci = v;
#pragma unroll
      for (int j = 0; j < 16; ++j) {
        if (cs > val[j]) {
          float t1 = val[j]; val[j] = cs; cs = t1;
          float t2 = lf[j];  lf[j] = cf; cf = t2;
          int   t3 = tk[j];  tk[j] = ci; ci = t3;
        }
      }
    }
  }

  __shared__ float sv[256 * 16];
  __shared__ float sl[256 * 16];
  __shared__ int   st[256 * 16];
#pragma unroll
  for (int j = 0; j < 16; ++j) { sv[tid*16+j] = val[j]; sl[tid*16+j] = lf[j]; st[tid*16+j] = tk[j]; }

  for (int srd = 128; srd >= 1; srd >>= 1) {
    __syncthreads();
    if (tid < srd) {
      int p = (tid + srd) * 16;
#pragma unroll
      for (int j = 0; j < 16; ++j) {
        float cs = sv[p + j], cf = sl[p + j]; int ci = st[p + j];
        if (cs > val[15] || (cs == val[15] && ci < tk[15])) {
#pragma unroll
          for (int k = 0; k < 16; ++k) {
            bool take = (cs > val[k]) || (cs == val[k] && ci < tk[k]);
            if (take) {
              float t1 = val[k]; val[k] = cs; cs = t1;
              float t2 = lf[k];  lf[k] = cf; cf = t2;
              int   t3 = tk[k];  tk[k] = ci; ci = t3;
            }
          }
        }
      }
#pragma unroll
      for (int j = 0; j < 16; ++j) { sv[tid*16+j] = val[j]; sl[tid*16+j] = lf[j]; st[tid*16+j] = tk[j]; }
    }
  }
  if (tid == 0) {
#pragma unroll
    for (int j = 0; j < 16; ++j) { cv[b*16+j] = val[j]; cl[b*16+j] = lf[j]; ctk[b*16+j] = tk[j]; }
  }
}

// ---- global merge + beam bookkeeping (thread 0) + embed/h/c gather (all 256 threads) ----
__global__ __launch_bounds__(256) void k_merge(
    const float* __restrict__ cv, const float* __restrict__ cl, const int* __restrict__ ctk,
    const int* __restrict__ seq_rd, int* __restrict__ seq_wr,
    const float* __restrict__ slp_rd, float* __restrict__ slp_wr,
    float* __restrict__ lp_sum,
    int* __restrict__ snap_seq, float* __restrict__ snap_lp, float* __restrict__ snap_p,
    const float* __restrict__ embed, const float* __restrict__ h, const float* __restrict__ c,
    float* __restrict__ A, float* __restrict__ cg, int t) {
  __shared__ int sq[16], stok[16];
  int tid = threadIdx.x;
  if (tid == 0) {
    unsigned char used[256];
    for (int i = 0; i < 256; ++i) used[i] = 0;
    float vals[16], lpfs[16]; int qb[16], tkk[16];
    for (int m = 0; m < 16; ++m) {
      float best = -3.4e38f; int bi = 0;
      for (int i = 0; i < 256; ++i)
        if (!used[i] && cv[i] > best) { best = cv[i]; bi = i; }
      used[bi] = 1;
      vals[m] = best; qb[m] = bi >> 4; tkk[m] = ctk[bi]; lpfs[m] = cl[bi];
    }
    int   ns[SEQ_LEN][BEAM];
    float nl[SEQ_LEN][BEAM];
    for (int m = 0; m < 16; ++m) {
      int q = qb[m];
      for (int s = 0; s < t; ++s) { ns[s][m] = seq_rd[s*BEAM+q]; nl[s][m] = slp_rd[s*BEAM+q]; }
      ns[t][m] = tkk[m]; nl[t][m] = lpfs[m];
      for (int s = t + 1; s < SEQ_LEN; ++s) { ns[s][m] = 0; nl[s][m] = 0.0f; }
    }
    for (int s = 0; s < SEQ_LEN; ++s)
      for (int m = 0; m < 16; ++m) { seq_wr[s*BEAM+m] = ns[s][m]; slp_wr[s*BEAM+m] = nl[s][m]; }
    for (int m = 0; m < 16; ++m) {
      bool fin = (tkk[m] == 0) || (t == SEQ_LEN - 1);
      snap_p[t*BEAM+m] = fin ? vals[m] : NEGV;
      for (int s = 0; s < SEQ_LEN; ++s) {
        snap_seq[(t*BEAM+m)*SEQ_LEN+s] = ns[s][m];
        snap_lp [(t*BEAM+m)*SEQ_LEN+s] = nl[s][m];
      }
      lp_sum[m] = fin ? -1000.0f : vals[m];
      sq[m] = qb[m]; stok[m] = tkk[m];
    }
  }
  __syncthreads();
  // gather A = [embed[tok] | h[q]], cg = c[q]   (16 x 1024 each)
  for (int i = tid; i < BEAM * RNN; i += 256) {
    int m = i >> 10, k = i & 1023;
    int tk = stok[m], q = sq[m];
    A[m*2048 + k]        = embed[(size_t)tk * EMB + k];
    A[m*2048 + 1024 + k] = h[q*RNN + k];
    cg[m*RNN + k]        = c[q*RNN + k];
  }
}

// ---------------- gates GEMM: g = A @ [Wih;Whh] + b  via wmma f32 16x16x4 ----------------
__global__ __launch_bounds__(32) void k_gates(const float* __restrict__ A,
                                              const float* __restrict__ Wih,
                                              const float* __restrict__ Whh,
                                              const float* __restrict__ bias,
                                              float* __restrict__ g) {
  int n0 = blockIdx.x * 16;
  int lane = threadIdx.x;
  int hf = lane >> 4, l = lane & 15;
  v8f acc = {};
  for (int k0 = 0; k0 < 2048; k0 += 4) {
    v2f a = *(const v2f*)(A + l*2048 + k0 + 2*hf);
    const float* Bp = (k0 < 1024) ? (Wih + (size_t)k0 * GATES)
                                  : (Whh + (size_t)(k0 - 1024) * GATES);
    v2f bb;
    bb.x = Bp[(size_t)(2*hf + 0) * GATES + n0 + l];
    bb.y = Bp[(size_t)(2*hf + 1) * GATES + n0 + l];
    acc = __builtin_amdgcn_wmma_f32_16x16x4_f32(false, a, false, bb, (short)0, acc, false, false);
  }
  int col = n0 + l;
  float bv = bias[col];
#pragma unroll
  for (int r = 0; r < 8; ++r) {
    int row = r + 8 * hf;
    g[row * GATES + col] = acc[r] + bv;
  }
}

// ---------------- LSTM cell elementwise ----------------
__global__ __launch_bounds__(256) void k_lstm(const float* __restrict__ g, const float* __restrict__ cg,
                                              float* __restrict__ h, float* __restrict__ c) {
  int i = blockIdx.x * blockDim.x + threadIdx.x;  // 16384
  int m = i >> 10, j = i & 1023;
  float i_ = g[m*GATES + j];
  float f_ = g[m*GATES + 1024 + j];
  float g_ = g[m*GATES + 2048 + j];
  float o_ = g[m*GATES + 3072 + j];
  float cn = sigm(f_) * cg[i] + sigm(i_) * tanhf(g_);
  c[i] = cn;
  h[i] = sigm(o_) * tanhf(cn);
}

// ---------------- output GEMM: logits = h @ Wout + bout ----------------
__global__ __launch_bounds__(32) void k_out(const float* __restrict__ h,
                                            const float* __restrict__ Wout,
                                            const float* __restrict__ bout,
                                            float* __restrict__ logits) {
  int n0 = blockIdx.x * 16;
  int lane = threadIdx.x;
  int hf = lane >> 4, l = lane & 15;
  int col = n0 + l;
  int scol = (col < VOCAB) ? col : (VOCAB - 1);   // clamp tail-tile loads in-bounds
  v8f acc = {};
  for (int k0 = 0; k0 < 1024; k0 += 4) {
    v2f a = *(const v2f*)(h + l*RNN + k0 + 2*hf);
    v2f bb;
    bb.x = Wout[(size_t)(k0 + 2*hf + 0) * VOCAB + scol];
    bb.y = Wout[(size_t)(k0 + 2*hf + 1) * VOCAB + scol];
    acc = __builtin_amdgcn_wmma_f32_16x16x4_f32(false, a, false, bb, (short)0, acc, false, false);
  }
  if (col < VOCAB) {
    float bv = bout[col];
#pragma unroll
    for (int r = 0; r < 8; ++r) {
      int row = r + 8 * hf;
      logits[(size_t)row * LSTRIDE + col] = acc[r] + bv;
    }
  }
}

// ---------------- final: top-16 over 320 done-beam snapshots -> outputs ----------------
__global__ void k_final(const int* __restrict__ snap_seq, const float* __restrict__ snap_lp,
                        const float* __restrict__ snap_p, float* __restrict__ out) {
  if (threadIdx.x != 0 || blockIdx.x != 0) return;
  unsigned char used[SEQ_LEN * BEAM];
  for (int i = 0; i < SEQ_LEN * BEAM; ++i) used[i] = 0;
  for (int m = 0; m < BEAM; ++m) {
    float best = -3.4e38f; int bi = 0;
    for (int i = 0; i < SEQ_LEN * BEAM; ++i)
      if (!used[i] && snap_p[i] > best) { best = snap_p[i]; bi = i; }
    used[bi] = 1;
    out[640 + m] = best;
    for (int s = 0; s < SEQ_LEN; ++s) {
      out[m * SEQ_LEN + s]       = (float)snap_seq[bi * SEQ_LEN + s];
      out[320 + m * SEQ_LEN + s] = snap_lp[bi * SEQ_LEN + s];
    }
  }
}

extern "C" void kernel_launch(void* const* d_in, const int* in_sizes, int n_in,
                              void* d_out, int out_size, void* d_ws, size_t ws_size,
                              hipStream_t stream) {
  const float* init_h   = (const float*)d_in[0];
  const float* init_c   = (const float*)d_in[1];
  const float* init_lp  = (const float*)d_in[2];
  const float* embed    = (const float*)d_in[3];
  const float* W_ih     = (const float*)d_in[4];
  const float* W_hh     = (const float*)d_in[5];
  const float* bvec     = (const float*)d_in[6];
  const float* W_out    = (const float*)d_in[7];
  const float* b_out    = (const float*)d_in[8];

  float* ws = (float*)d_ws;
  float* logits  = ws + OFF_LOGITS;
  float* A       = ws + OFF_A;
  float* g       = ws + OFF_G;
  float* h       = ws + OFF_H;
  float* c       = ws + OFF_C;
  float* cg      = ws + OFF_CG;
  float* lp_sum  = ws + OFF_LPSUM;
  float* cv      = ws + OFF_CVAL;
  float* cl      = ws + OFF_CLPF;
  int*   ctk     = (int*)(ws + OFF_CTOK);
  int*   seq0    = (int*)(ws + OFF_SEQ0);
  int*   seq1    = (int*)(ws + OFF_SEQ1);
  float* slp0    = ws + OFF_SLP0;
  float* slp1    = ws + OFF_SLP1;
  int*   snapseq = (int*)(ws + OFF_SNAPSEQ);
  float* snaplp  = ws + OFF_SNAPLP;
  float* snapp   = ws + OFF_SNAPP;

  k_init<<<64, 256, 0, stream>>>(init_h, init_c, h, c, lp_sum);

  for (int t = 0; t < SEQ_LEN; ++t) {
    const float* src = (t == 0) ? init_lp : logits;
    int stride = (t == 0) ? VOCAB : LSTRIDE;
    int*   seq_rd = (t & 1) ? seq1 : seq0;
    int*   seq_wr = (t & 1) ? seq0 : seq1;
    float* slp_rd = (t & 1) ? slp1 : slp0;
    float* slp_wr = (t & 1) ? slp0 : slp1;

    k_topk<<<BEAM, 256, 0, stream>>>(src, stride, seq_rd, lp_sum, t, cv, cl, ctk);
    k_merge<<<1, 256, 0, stream>>>(cv, cl, ctk, seq_rd, seq_wr, slp_rd, slp_wr,
                                   lp_sum, snapseq, snaplp, snapp,
                                   embed, h, c, A, cg, t);
    if (t < SEQ_LEN - 1) {
      k_gates<<<NTILES_G, 32, 0, stream>>>(A, W_ih, W_hh, bvec, g);
      k_lstm<<<64, 256, 0, stream>>>(g, cg, h, c);
      k_out<<<NTILES_OUT, 32, 0, stream>>>(h, W_out, b_out, logits);
    }
  }
  k_final<<<1, 32, 0, stream>>>(snapseq, snaplp, snapp, (float*)d_out);
}